// GIN_23364622090833
// MI455X (gfx1250) — compile-verified
//
#include <hip/hip_runtime.h>
#include <hip/hip_bf16.h>

#define N_NODES   100000
#define N_EDGES   3200000
#define IN_CH     128
#define HID       64
#define N_CLASSES 10

typedef __attribute__((ext_vector_type(2))) float v2f;
typedef __attribute__((ext_vector_type(8))) float v8f;

// ---------------------------------------------------------------------------
// Zero-fill (grid-stride, float4)
// ---------------------------------------------------------------------------
__global__ void zero_kernel(float4* __restrict__ p, long n4) {
    long i      = (long)blockIdx.x * blockDim.x + threadIdx.x;
    long stride = (long)gridDim.x * blockDim.x;
    float4 z = make_float4(0.f, 0.f, 0.f, 0.f);
    for (; i < n4; i += stride) p[i] = z;
}

// ---------------------------------------------------------------------------
// Edge scatter-add: agg[dst[e]] += feat[src[e]]   (C channels, float4 lanes)
// C/4 consecutive threads cooperate on one edge -> fully coalesced 4B*C bursts
// for both the gather and the atomic scatter. agg fits in the 192MB L2, so
// the fp32 atomics are L2-resident RMWs.
// ---------------------------------------------------------------------------
template <int C>
__global__ void scatter_kernel(const float* __restrict__ feat,
                               const int*   __restrict__ src,
                               const int*   __restrict__ dst,
                               float*       __restrict__ agg) {
    constexpr int LPE = C / 4;                 // lanes per edge
    long tid   = (long)blockIdx.x * blockDim.x + threadIdx.x;
    long total = (long)N_EDGES * LPE;
    if (tid >= total) return;
    int e   = (int)(tid / LPE);
    int sub = (int)(tid % LPE);
    int s = src[e];
    int d = dst[e];
    float4 v = ((const float4*)(feat + (long)s * C))[sub];
    float* out = agg + (long)d * C + sub * 4;
    unsafeAtomicAdd(out + 0, v.x);
    unsafeAtomicAdd(out + 1, v.y);
    unsafeAtomicAdd(out + 2, v.z);
    unsafeAtomicAdd(out + 3, v.w);
}

// ---------------------------------------------------------------------------
// WMMA helper note (V_WMMA_F32_16X16X4_F32, wave32):
//  A (16x4, M x K): lanes 0-15 -> M=lane, VGPR{0,1} = K{0,1};
//                   lanes 16-31 -> M=lane-16, VGPR{0,1} = K{2,3}.
//  B (4x16, K x N): lanes 0-15 -> N=lane, VGPR{0,1} = K{0,1};
//                   lanes 16-31 -> N=lane-16, VGPR{0,1} = K{2,3}.
//  C/D (16x16):     VGPR v: lanes 0-15 -> (M=v, N=lane),
//                           lanes 16-31 -> (M=v+8, N=lane-16).
// ---------------------------------------------------------------------------

// ---------------------------------------------------------------------------
// MLP of GIN layer 1 (fused):
//   h1 = relu( relu( (agg1 + x) @ W1a + b1a ) @ W2a + b2a )
// Block = 128 threads (4 waves), tile = 64 rows; each wave owns 16 rows.
// ---------------------------------------------------------------------------
__global__ void __launch_bounds__(128)
gin_mlp1_kernel(const float* __restrict__ x,   const float* __restrict__ agg,
                const float* __restrict__ W1,  const float* __restrict__ b1,
                const float* __restrict__ W2,  const float* __restrict__ b2,
                float* __restrict__ h_out) {
    __shared__ float xs[64][IN_CH + 4];   // (agg + x) tile, padded vs bank conflicts
    __shared__ float w1s[IN_CH][HID];
    __shared__ float w2s[HID][HID];
    __shared__ float ts[64][HID + 4];     // relu(gemm1) intermediate
    __shared__ float b1s[HID], b2s[HID];

    const int  tid = threadIdx.x;
    const long m0  = (long)blockIdx.x * 64;

    // --- cooperative staging ------------------------------------------------
    for (int i = tid; i < 64 * (IN_CH / 4); i += 128) {
        int  r  = i / (IN_CH / 4);
        int  c4 = i % (IN_CH / 4);
        long row = m0 + r; if (row >= N_NODES) row = N_NODES - 1;  // clamp tail
        float4 a = ((const float4*)(agg + row * IN_CH))[c4];
        float4 b = ((const float4*)(x   + row * IN_CH))[c4];
        ((float4*)&xs[r][0])[c4] = make_float4(a.x + b.x, a.y + b.y,
                                               a.z + b.z, a.w + b.w);
    }
    for (int i = tid; i < IN_CH * HID / 4; i += 128)
        ((float4*)&w1s[0][0])[i] = ((const float4*)W1)[i];
    for (int i = tid; i < HID * HID / 4; i += 128)
        ((float4*)&w2s[0][0])[i] = ((const float4*)W2)[i];
    if (tid < HID) { b1s[tid] = b1[tid]; b2s[tid] = b2[tid]; }
    __syncthreads();

    const int wave = tid >> 5;
    const int lane = tid & 31;
    const int m    = lane & 15;      // row within wave tile / B column
    const int kh   = lane >> 4;      // K-half select (0 -> K0,1 ; 1 -> K2,3)
    const int rb   = wave * 16;      // wave's row base inside block tile

    // --- GEMM1: ts = relu(xs @ W1 + b1), K = 128 ---------------------------
    #pragma unroll
    for (int nt = 0; nt < 4; ++nt) {
        const int n = nt * 16 + m;
        v8f acc = {};
        #pragma unroll 8
        for (int kb = 0; kb < IN_CH; kb += 4) {
            const int kk = kb + 2 * kh;
            v2f a = *(const v2f*)&xs[rb + m][kk];     // ds_load_b64
            v2f b; b.x = w1s[kk][n]; b.y = w1s[kk + 1][n];
            acc = __builtin_amdgcn_wmma_f32_16x16x4_f32(
                false, a, false, b, (short)0, acc, false, false);
        }
        #pragma unroll
        for (int v = 0; v < 8; ++v) {
            float t = acc[v] + b1s[n];
            ts[rb + v + 8 * kh][n] = t > 0.f ? t : 0.f;
        }
    }
    // same-wave LDS ops are in-order; each wave reads only its own 16 rows.

    // --- GEMM2: h1 = relu(ts @ W2 + b2), K = 64 ----------------------------
    #pragma unroll
    for (int nt = 0; nt < 4; ++nt) {
        const int n = nt * 16 + m;
        v8f acc = {};
        #pragma unroll 8
        for (int kb = 0; kb < HID; kb += 4) {
            const int kk = kb + 2 * kh;
            v2f a = *(const v2f*)&ts[rb + m][kk];
            v2f b; b.x = w2s[kk][n]; b.y = w2s[kk + 1][n];
            acc = __builtin_amdgcn_wmma_f32_16x16x4_f32(
                false, a, false, b, (short)0, acc, false, false);
        }
        #pragma unroll
        for (int v = 0; v < 8; ++v) {
            long grow = m0 + rb + v + 8 * kh;
            if (grow < N_NODES) {
                float t = acc[v] + b2s[n];
                h_out[grow * HID + n] = t > 0.f ? t : 0.f;  // inter-layer relu
            }
        }
    }
}

// ---------------------------------------------------------------------------
// MLP of GIN layer 2 + classifier head (fused):
//   h2     = relu( relu( (agg2 + h1) @ W1b + b1b ) @ W2b + b2b )
//   out    = log_softmax( h2 @ Wl + bl )
// ---------------------------------------------------------------------------
__global__ void __launch_bounds__(128)
gin_mlp2_kernel(const float* __restrict__ h_in, const float* __restrict__ agg,
                const float* __restrict__ W1,   const float* __restrict__ b1,
                const float* __restrict__ W2,   const float* __restrict__ b2,
                const float* __restrict__ Wl,   const float* __restrict__ bl,
                float* __restrict__ out) {
    __shared__ float xs[64][HID + 4];     // (agg2 + h1) tile
    __shared__ float w1s[HID][HID];
    __shared__ float w2s[HID][HID];
    __shared__ float ts[64][HID + 4];     // relu(gemm1)
    __shared__ float hs[64][HID + 4];     // h2 tile for the head
    __shared__ float wls[HID][N_CLASSES];
    __shared__ float b1s[HID], b2s[HID], bls[N_CLASSES];

    const int  tid = threadIdx.x;
    const long m0  = (long)blockIdx.x * 64;

    for (int i = tid; i < 64 * (HID / 4); i += 128) {
        int  r  = i / (HID / 4);
        int  c4 = i % (HID / 4);
        long row = m0 + r; if (row >= N_NODES) row = N_NODES - 1;
        float4 a = ((const float4*)(agg  + row * HID))[c4];
        float4 b = ((const float4*)(h_in + row * HID))[c4];
        ((float4*)&xs[r][0])[c4] = make_float4(a.x + b.x, a.y + b.y,
                                               a.z + b.z, a.w + b.w);
    }
    for (int i = tid; i < HID * HID / 4; i += 128) {
        ((float4*)&w1s[0][0])[i] = ((const float4*)W1)[i];
        ((float4*)&w2s[0][0])[i] = ((const float4*)W2)[i];
    }
    for (int i = tid; i < HID * N_CLASSES; i += 128)
        (&wls[0][0])[i] = Wl[i];
    if (tid < HID) { b1s[tid] = b1[tid]; b2s[tid] = b2[tid]; }
    if (tid < N_CLASSES) bls[tid] = bl[tid];
    __syncthreads();

    const int wave = tid >> 5;
    const int lane = tid & 31;
    const int m    = lane & 15;
    const int kh   = lane >> 4;
    const int rb   = wave * 16;

    // --- GEMM1 (K = 64) ----------------------------------------------------
    #pragma unroll
    for (int nt = 0; nt < 4; ++nt) {
        const int n = nt * 16 + m;
        v8f acc = {};
        #pragma unroll 8
        for (int kb = 0; kb < HID; kb += 4) {
            const int kk = kb + 2 * kh;
            v2f a = *(const v2f*)&xs[rb + m][kk];
            v2f b; b.x = w1s[kk][n]; b.y = w1s[kk + 1][n];
            acc = __builtin_amdgcn_wmma_f32_16x16x4_f32(
                false, a, false, b, (short)0, acc, false, false);
        }
        #pragma unroll
        for (int v = 0; v < 8; ++v) {
            float t = acc[v] + b1s[n];
            ts[rb + v + 8 * kh][n] = t > 0.f ? t : 0.f;
        }
    }

    // --- GEMM2 (K = 64) -> h2 tile in LDS ----------------------------------
    #pragma unroll
    for (int nt = 0; nt < 4; ++nt) {
        const int n = nt * 16 + m;
        v8f acc = {};
        #pragma unroll 8
        for (int kb = 0; kb < HID; kb += 4) {
            const int kk = kb + 2 * kh;
            v2f a = *(const v2f*)&ts[rb + m][kk];
            v2f b; b.x = w2s[kk][n]; b.y = w2s[kk + 1][n];
            acc = __builtin_amdgcn_wmma_f32_16x16x4_f32(
                false, a, false, b, (short)0, acc, false, false);
        }
        #pragma unroll
        for (int v = 0; v < 8; ++v) {
            float t = acc[v] + b2s[n];
            hs[rb + v + 8 * kh][n] = t > 0.f ? t : 0.f;
        }
    }
    __syncthreads();

    // --- head: logits (64x10 GEMV) + log_softmax, one row per thread -------
    if (tid < 64) {
        long grow = m0 + tid;
        if (grow < N_NODES) {
            float logit[N_CLASSES];
            #pragma unroll
            for (int c = 0; c < N_CLASSES; ++c) logit[c] = bls[c];
            for (int k = 0; k < HID; ++k) {
                float hv = hs[tid][k];
                #pragma unroll
                for (int c = 0; c < N_CLASSES; ++c)
                    logit[c] = fmaf(hv, wls[k][c], logit[c]);
            }
            float mx = logit[0];
            #pragma unroll
            for (int c = 1; c < N_CLASSES; ++c) mx = fmaxf(mx, logit[c]);
            float se = 0.f;
            #pragma unroll
            for (int c = 0; c < N_CLASSES; ++c) se += expf(logit[c] - mx);
            float lse = mx + logf(se);
            #pragma unroll
            for (int c = 0; c < N_CLASSES; ++c)
                out[grow * N_CLASSES + c] = logit[c] - lse;
        }
    }
}

// ---------------------------------------------------------------------------
// Launch
// ---------------------------------------------------------------------------
extern "C" void kernel_launch(void* const* d_in, const int* in_sizes, int n_in,
                              void* d_out, int out_size, void* d_ws, size_t ws_size,
                              hipStream_t stream) {
    const float* x   = (const float*)d_in[0];
    const int*   ei  = (const int*)  d_in[1];
    const float* W1a = (const float*)d_in[2];
    const float* b1a = (const float*)d_in[3];
    const float* W2a = (const float*)d_in[4];
    const float* b2a = (const float*)d_in[5];
    const float* W1b = (const float*)d_in[6];
    const float* b1b = (const float*)d_in[7];
    const float* W2b = (const float*)d_in[8];
    const float* b2b = (const float*)d_in[9];
    const float* Wl  = (const float*)d_in[10];
    const float* bl  = (const float*)d_in[11];
    float* out = (float*)d_out;

    // workspace layout: agg1 (N*128) | agg2 (N*64) | h1 (N*64)  -- ~102.4 MB
    float* agg1 = (float*)d_ws;
    float* agg2 = agg1 + (long)N_NODES * IN_CH;
    float* h1   = agg2 + (long)N_NODES * HID;

    const int* src = ei;             // edge_index[0]
    const int* dst = ei + N_EDGES;   // edge_index[1]

    // 1) zero agg1 + agg2 (contiguous span)
    long n4 = (long)N_NODES * (IN_CH + HID) / 4;
    zero_kernel<<<4096, 256, 0, stream>>>((float4*)agg1, n4);

    // 2) layer-1 aggregation: agg1[d] += x[s]
    {
        long t = (long)N_EDGES * (IN_CH / 4);
        scatter_kernel<IN_CH><<<(int)((t + 255) / 256), 256, 0, stream>>>(
            x, src, dst, agg1);
    }

    // 3) fused MLP1 -> h1 (includes inter-layer relu)
    int nblk = (N_NODES + 63) / 64;
    gin_mlp1_kernel<<<nblk, 128, 0, stream>>>(x, agg1, W1a, b1a, W2a, b2a, h1);

    // 4) layer-2 aggregation: agg2[d] += h1[s]
    {
        long t = (long)N_EDGES * (HID / 4);
        scatter_kernel<HID><<<(int)((t + 255) / 256), 256, 0, stream>>>(
            h1, src, dst, agg2);
    }

    // 5) fused MLP2 + classifier + log_softmax -> out
    gin_mlp2_kernel<<<nblk, 128, 0, stream>>>(h1, agg2, W1b, b1b, W2b, b2b,
                                              Wl, bl, out);
}